// MorphPipeline_13271448945088
// MI455X (gfx1250) — compile-verified
//
#include <hip/hip_runtime.h>
#include <math.h>

// CDNA5 / gfx1250: wave32, FP32 WMMA 16x16x4.
typedef __attribute__((ext_vector_type(2))) float v2f;
typedef __attribute__((ext_vector_type(8))) float v8f;

union V8 { v8f v; float f[8]; };

struct Rotor { float c, b3, b5, b6; };   // nonzero comps of a Cl(3,0) rotor: e0, e12, e13, e23

// sign from reordering basis blades a*b into canonical order (3-bit closed form)
__device__ __forceinline__ float reorder_sign(int a, int b) {
    int s = __popc((a >> 1) & b) + __popc((a >> 2) & b);
    return (s & 1) ? -1.0f : 1.0f;
}

__device__ __forceinline__ float rotor_comp(const Rotor& r, int idx) {
    float v = 0.0f;
    v = (idx == 0) ? r.c  : v;
    v = (idx == 3) ? r.b3 : v;
    v = (idx == 5) ? r.b5 : v;
    v = (idx == 6) ? r.b6 : v;
    return v;   // v_cndmask chain, no scratch
}

// gp(R, x)_m = sum_k Lmat[m][k] x_k,  Lmat[m][k] = R[m^k] * sign(m^k, k)
__device__ __forceinline__ float lmat_elem(const Rotor& r, int m, int k) {
    return rotor_comp(r, m ^ k) * reorder_sign(m ^ k, k);
}
// gp(y, Rr)_m = sum_k Rmat[m][k] y_k,  Rmat[m][k] = Rr[k^m] * sign(k, k^m)
__device__ __forceinline__ float rmat_elem(const Rotor& r, int m, int k) {
    return rotor_comp(r, m ^ k) * reorder_sign(k, m ^ k);
}

// exact exp of simple bivector: R = cos(th) + sin(th)/th * (-B/2)
// th is tiny here (inputs scaled by 0.01), so hardware v_sin/v_cos are safe
// and avoid the ~100x larger libm argument-reduction sequences.
__device__ __forceinline__ void make_rotor(float b0, float b1, float b2,
                                           Rotor& R, Rotor& Rrev) {
    float h0 = -0.5f * b0, h1 = -0.5f * b1, h2 = -0.5f * b2;
    float t2 = h0 * h0 + h1 * h1 + h2 * h2;
    float th = __builtin_sqrtf(t2);
    float sc = (th > 1e-6f) ? (__sinf(th) / th) : (1.0f - t2 * (1.0f / 6.0f));
    R.c = __cosf(th); R.b3 = sc * h0; R.b5 = sc * h1; R.b6 = sc * h2;
    Rrev.c = R.c; Rrev.b3 = -R.b3; Rrev.b5 = -R.b5; Rrev.b6 = -R.b6;
}

// A-matrix tile (16x4 f32): lane L, VGPR v holds A[L&15][4c + v + (L>=16 ? 2 : 0)]
// Branchless: rows 8..15 of the padded 16x16 selected to zero (no EXEC juggling).
__device__ __forceinline__ v2f buildA(const Rotor& r, bool right, int mrow, int hi, int c) {
    int k0 = 4 * c + (hi ? 2 : 0);
    float a0 = right ? rmat_elem(r, mrow, k0)     : lmat_elem(r, mrow, k0);
    float a1 = right ? rmat_elem(r, mrow, k0 + 1) : lmat_elem(r, mrow, k0 + 1);
    v2f a;
    a.x = (mrow < 8) ? a0 : 0.0f;
    a.y = (mrow < 8) ? a1 : 0.0f;
    return a;
}

__device__ __forceinline__ void zero8(V8& a) {
#pragma unroll
    for (int i = 0; i < 8; ++i) a.f[i] = 0.0f;
}

// D (C/D layout: VGPR r = row r lanes 0-15, row r+8 lanes 16-31)
//   -> B (VGPR v = row 4c+v lanes 0-15, row 4c+v+2 lanes 16-31), via cross-lane shuffle.
__device__ __forceinline__ void relayout(const V8& acc, int mrow, int hi, v2f b[2]) {
#pragma unroll
    for (int c = 0; c < 2; ++c) {
        float e[2];
#pragma unroll
        for (int v = 0; v < 2; ++v) {
            float lo = __shfl(acc.f[4 * c + v],     mrow, 32);
            float hh = __shfl(acc.f[4 * c + v + 2], mrow, 32);
            e[v] = hi ? hh : lo;
        }
        b[c].x = e[0]; b[c].y = e[1];
    }
}

// keep only column `node` (value-level mask; EXEC untouched so WMMA stays legal)
__device__ __forceinline__ v2f maskB(v2f b, int ncol, int node) {
    v2f z;
    z.x = (ncol == node) ? b.x : 0.0f;
    z.y = (ncol == node) ? b.y : 0.0f;
    return z;
}

__device__ __forceinline__ v8f wmma4(v2f a, v2f b, v8f c) {
    // (neg_a, A, neg_b, B, c_mod, C, reuse_a, reuse_b)
    return __builtin_amdgcn_wmma_f32_16x16x4_f32(false, a, false, b, (short)0, c,
                                                 false, false);
}

__global__ void __launch_bounds__(32)
morph_pipeline_kernel(const float* __restrict__ basis,   // (3, 8)
                      const float* __restrict__ gbv,     // (3, 3)
                      const float* __restrict__ tbv,     // (3, 3, 3)
                      const float* __restrict__ ls,      // (3, 3)
                      float* __restrict__ out)           // 24 (final) + 72 (intermediates)
{
    const int lane = threadIdx.x;
    const int mrow = lane & 15;
    const int hi   = lane >> 4;
    const int ncol = mrow;          // column index in B/C/D layouts

    // Initial B operand: X[k][n] = basis[n*8 + k], columns 0..2, rows 0..7; rest zero.
    v2f b[2];
#pragma unroll
    for (int c = 0; c < 2; ++c) {
        int k0 = 4 * c + (hi ? 2 : 0);
        b[c].x = (ncol < 3) ? basis[ncol * 8 + k0]     : 0.0f;
        b[c].y = (ncol < 3) ? basis[ncol * 8 + k0 + 1] : 0.0f;
    }

    // Fully unrolled: lets the scheduler hoist all uniform s_loads of stage
    // parameters to the top and hide them under stage-0 WMMA work.
#pragma unroll
    for (int s = 0; s < 3; ++s) {
        Rotor Rg, Rgrev;
        make_rotor(gbv[s * 3 + 0], gbv[s * 3 + 1], gbv[s * 3 + 2], Rg, Rgrev);

        V8 acc;

        // ---- global rotor sandwich: X <- Rmat(Rg~) @ (Lmat(Rg) @ X) ----
        zero8(acc);
        acc.v = wmma4(buildA(Rg, false, mrow, hi, 0), b[0], acc.v);
        acc.v = wmma4(buildA(Rg, false, mrow, hi, 1), b[1], acc.v);
        relayout(acc, mrow, hi, b);

        zero8(acc);
        acc.v = wmma4(buildA(Rgrev, true, mrow, hi, 0), b[0], acc.v);
        acc.v = wmma4(buildA(Rgrev, true, mrow, hi, 1), b[1], acc.v);
        relayout(acc, mrow, hi, b);

        // ---- per-node twist, left halves: Y[:,i] = Lmat(T_i) @ X[:,i] ----
        zero8(acc);
#pragma unroll
        for (int node = 0; node < 3; ++node) {
            Rotor T, Trev;
            make_rotor(tbv[s * 9 + node * 3 + 0], tbv[s * 9 + node * 3 + 1],
                       tbv[s * 9 + node * 3 + 2], T, Trev);
            acc.v = wmma4(buildA(T, false, mrow, hi, 0), maskB(b[0], ncol, node), acc.v);
            acc.v = wmma4(buildA(T, false, mrow, hi, 1), maskB(b[1], ncol, node), acc.v);
        }
        relayout(acc, mrow, hi, b);

        // ---- per-node twist, right halves: Z[:,i] = Rmat(T_i~) @ Y[:,i] ----
        zero8(acc);
#pragma unroll
        for (int node = 0; node < 3; ++node) {
            Rotor T, Trev;
            make_rotor(tbv[s * 9 + node * 3 + 0], tbv[s * 9 + node * 3 + 1],
                       tbv[s * 9 + node * 3 + 2], T, Trev);
            acc.v = wmma4(buildA(Trev, true, mrow, hi, 0), maskB(b[0], ncol, node), acc.v);
            acc.v = wmma4(buildA(Trev, true, mrow, hi, 1), maskB(b[1], ncol, node), acc.v);
        }

        // ---- dynamic scale per node (column): exp(clip(ls, -3, 3)) ----
        float lsv = (ncol < 3) ? ls[s * 3 + ncol] : 0.0f;
        lsv = fminf(3.0f, fmaxf(-3.0f, lsv));
        float scale = __expf(lsv);    // |arg| <= 3: hardware v_exp path is exact enough
#pragma unroll
        for (int r = 0; r < 8; ++r) acc.f[r] *= scale;

        // ---- store intermediates: lanes 0..2 hold columns 0..2, rows 0..7 ----
        if (hi == 0 && ncol < 3) {
#pragma unroll
            for (int k = 0; k < 8; ++k)
                out[24 + s * 24 + ncol * 8 + k] = acc.f[k];
            if (s == 2) {
#pragma unroll
                for (int k = 0; k < 8; ++k)
                    out[ncol * 8 + k] = acc.f[k];   // final x
            }
        }

        if (s < 2) relayout(acc, mrow, hi, b);   // feed next stage
    }
}

extern "C" void kernel_launch(void* const* d_in, const int* in_sizes, int n_in,
                              void* d_out, int out_size, void* d_ws, size_t ws_size,
                              hipStream_t stream) {
    (void)in_sizes; (void)n_in; (void)out_size; (void)d_ws; (void)ws_size;
    const float* basis = (const float*)d_in[0];  // basis_mvs (3,8)
    const float* gbv   = (const float*)d_in[1];  // global_bv (3,3)
    const float* tbv   = (const float*)d_in[2];  // twist_bvs (3,3,3)
    const float* ls    = (const float*)d_in[3];  // log_scale (3,3)
    float* out = (float*)d_out;                  // 96 floats: x (24) ++ intermediates (72)
    // Latency-bound micro-problem: one wave32, one WGP, everything in registers.
    morph_pipeline_kernel<<<1, 32, 0, stream>>>(basis, gbv, tbv, ls, out);
}